// RectifiedFlowMatcher_30399778521475
// MI455X (gfx1250) — compile-verified
//
// RectifiedFlowMatcher for MI455X (gfx1250, wave32).
// Strategy: never materialize the 8x2048x2048 cost matrices. Point sets
// (2048 x (xyz,0) + |p|^2) live in LDS; every logsumexp sweep recomputes
// cost tiles with V_WMMA_F32_16X16X4_F32 (K=4 exactly fits 3-D points,
// fp32 keeps Sinkhorn exponents exact). MLP GEMMs use the same WMMA path.
// Logsumexp is branchless and in base-2 (v_exp_f32 native), with tile-level
// max batching to minimize transcendental count.
// Workspace use: ~1.5 MB of d_ws (assumed available).

#include <hip/hip_runtime.h>
#include <math.h>

typedef __attribute__((ext_vector_type(2))) float v2f;
typedef __attribute__((ext_vector_type(8))) float v8f;

#define B_SIZE   8
#define N_PTS    16384
#define NSUB     2048
#define DZ       256
#define DH       256
#define DIN      260       // 3 + 1 + 256
#define DIN_PAD  264       // padded to multiple of 4 (K steps)
#define S_ITERS  20
#define S_REG    0.05f
#define OT_EPS   0.0025f   // OT_BLUR^2
#define OT_ITERS 20
#define LOG_NU   (-7.6246189861593985f)   // -log(2048)
#define NEG_INF  (-__builtin_inff())
#define LOG2E    1.4426950408889634f
#define LN2      0.6931471805599453f

// ---------------------------------------------------------------- WMMA core
__device__ __forceinline__ v8f wmma_k4(v2f a, v2f b, v8f c) {
  // D(16x16,f32) = A(16x4,f32) * B(4x16,f32) + C
  return __builtin_amdgcn_wmma_f32_16x16x4_f32(
      /*neg_a=*/false, a, /*neg_b=*/false, b,
      /*c_mod=*/(short)0, c, /*reuse_a=*/false, /*reuse_b=*/false);
}

// Branchless base-2 logsumexp helpers -------------------------------------
// Accumulate a pre-reduced chunk (tm = chunk max, ts = sum exp2(arg-tm)).
__device__ __forceinline__ void lse2_accum(float& m, float& s, float tm, float ts) {
  float M = fmaxf(m, tm);
  s = s * exp2f(m - M) + ts * exp2f(tm - M);
  m = M;
}
__device__ __forceinline__ void lse2_merge(float& m, float& s, float m2, float s2) {
  float M = fmaxf(m, m2);
  s = s * exp2f(m - M) + s2 * exp2f(m2 - M);
  m = M;
}

// ------------------------------------------------------- template preprocess
__global__ __launch_bounds__(256) void prep_template_kernel(
    const float* __restrict__ tmpl, float* __restrict__ tpts,
    float* __restrict__ tsq, float* __restrict__ accs) {
  __shared__ float sr[256];
  int i = blockIdx.x * 256 + threadIdx.x;
  float d2 = 0.0f;
  if (i < NSUB) {
    float x = tmpl[i * 3 + 0], y = tmpl[i * 3 + 1], z = tmpl[i * 3 + 2];
    tpts[i * 4 + 0] = x; tpts[i * 4 + 1] = y; tpts[i * 4 + 2] = z;
    tpts[i * 4 + 3] = 0.0f;
    float sq = x * x + y * y + z * z;
    tsq[i] = sq;
    float n = sqrtf(sq) - 1.0f;
    d2 = n * n;
  }
  sr[threadIdx.x] = d2;
  __syncthreads();
  for (int off = 128; off > 0; off >>= 1) {
    if (threadIdx.x < off) sr[threadIdx.x] += sr[threadIdx.x + off];
    __syncthreads();
  }
  if (threadIdx.x == 0) atomicAdd(&accs[1], sr[0]);
}

// -------------------------------------------------- farthest point sampling
__global__ __launch_bounds__(1024) void fps_kernel(
    const float* __restrict__ target, float* __restrict__ tsubp,
    float* __restrict__ tsubsq) {
  __shared__ float sVal[32];
  __shared__ int   sIdx[32];
  __shared__ float sCent[4];
  __shared__ int   sFar;
  const int b = blockIdx.x, tid = threadIdx.x;
  const float* xb = target + (long)b * N_PTS * 3;
  float px[16], py[16], pz[16], dd[16];
#pragma unroll
  for (int j = 0; j < 16; ++j) {
    int idx = tid + 1024 * j;
    px[j] = xb[idx * 3 + 0];
    py[j] = xb[idx * 3 + 1];
    pz[j] = xb[idx * 3 + 2];
    dd[j] = __builtin_inff();
  }
  int far = 0;
  for (int i = 0; i < NSUB; ++i) {
    if (tid == (far & 1023)) {                 // owner of selected point
      int j = far >> 10;
      sCent[0] = px[j]; sCent[1] = py[j]; sCent[2] = pz[j];
      long o = ((long)b * NSUB + i) * 4;
      tsubp[o + 0] = px[j]; tsubp[o + 1] = py[j];
      tsubp[o + 2] = pz[j]; tsubp[o + 3] = 0.0f;
      tsubsq[(long)b * NSUB + i] = px[j]*px[j] + py[j]*py[j] + pz[j]*pz[j];
    }
    __syncthreads();
    float cx = sCent[0], cy = sCent[1], cz = sCent[2];
    float bv = -1.0f; int bi = 0;
#pragma unroll
    for (int j = 0; j < 16; ++j) {
      float dx = px[j]-cx, dy = py[j]-cy, dz = pz[j]-cz;
      float d = dx*dx + dy*dy + dz*dz;
      dd[j] = fminf(dd[j], d);
      int idx = tid + 1024 * j;
      bool better = (dd[j] > bv) || (dd[j] == bv && idx < bi);
      bv = better ? dd[j] : bv;
      bi = better ? idx   : bi;
    }
    for (int off = 16; off >= 1; off >>= 1) {   // wave argmax (branchless)
      float ov = __shfl_xor(bv, off, 32);
      int   oi = __shfl_xor(bi, off, 32);
      bool better = (ov > bv) || (ov == bv && oi < bi);
      bv = better ? ov : bv;
      bi = better ? oi : bi;
    }
    if ((tid & 31) == 0) { sVal[tid >> 5] = bv; sIdx[tid >> 5] = bi; }
    __syncthreads();
    if (tid < 32) {                             // cross-wave argmax
      float v = sVal[tid]; int ix = sIdx[tid];
      for (int off = 16; off >= 1; off >>= 1) {
        float ov = __shfl_xor(v, off, 32);
        int   oi = __shfl_xor(ix, off, 32);
        bool better = (ov > v) || (ov == v && oi < ix);
        v  = better ? ov : v;
        ix = better ? oi : ix;
      }
      if (tid == 0) sFar = ix;
    }
    __syncthreads();
    far = sFar;
  }
}

// --------------------------------------------- generic logsumexp row sweep
// out[n] = rscaleLn2 * lse2_m( alpha2*dist(X[n],Y[m]) + beta2*colv[m] + gamma2 )
// (all exponent constants pre-scaled by log2(e); result scaled by ln2)
__global__ __launch_bounds__(256) void lse_row_pass(
    const float* __restrict__ Xpts, const float* __restrict__ Xsq,
    int xPtsStride, int xSqStride,
    const float* __restrict__ Ypts, const float* __restrict__ Ysq,
    const float* __restrict__ colv, float* __restrict__ outRow,
    float alpha2, float beta2, float gamma2, float rscaleLn2) {
  __shared__ __align__(16) float sY[NSUB * 4];
  __shared__ float sYsq[NSUB];
  __shared__ float sCv[NSUB];    // beta2*colv + gamma2 (pre-folded)
  const int b = blockIdx.y;
  const float* xp = Xpts + (long)b * xPtsStride;
  const float* xs = Xsq  + (long)b * xSqStride;
  const float* yp = Ypts + (long)b * (NSUB * 4);
  const float* ys = Ysq  + (long)b * NSUB;
  const float* cv = colv + (long)b * NSUB;
  float* out = outRow + (long)b * NSUB;
  const int tid = threadIdx.x;
  for (int i = tid; i < NSUB; i += 256) {
    ((float4*)sY)[i] = ((const float4*)yp)[i];
    sYsq[i] = ys[i];
    sCv[i]  = beta2 * cv[i] + gamma2;
  }
  __syncthreads();
  const int wave = tid >> 5, lane = tid & 31, li = lane & 15;
  const bool hi = lane >= 16;
  const int q0 = hi ? 2 : 0, q1 = hi ? 3 : 1;
  const int rowbase = (blockIdx.x * 8 + wave) * 16;

  v2f a;                                       // A: 16 rows x K4
  a.x = xp[(rowbase + li) * 4 + q0];
  a.y = xp[(rowbase + li) * 4 + q1];
  float xsqv[8];
#pragma unroll
  for (int i = 0; i < 8; ++i) xsqv[i] = xs[rowbase + i + (hi ? 8 : 0)];

  float mm[8], ss[8];
#pragma unroll
  for (int i = 0; i < 8; ++i) { mm[i] = NEG_INF; ss[i] = 0.0f; }

  // 4 tiles in flight: 4 independent WMMAs; 6 exps per 4 elements per row.
  for (int t = 0; t < NSUB / 16; t += 4) {
    const int c0 = (t + 0) * 16 + li, c1 = (t + 1) * 16 + li;
    const int c2 = (t + 2) * 16 + li, c3 = (t + 3) * 16 + li;
    v2f b0, b1v, b2v, b3;
    b0.x  = sY[c0 * 4 + q0]; b0.y  = sY[c0 * 4 + q1];
    b1v.x = sY[c1 * 4 + q0]; b1v.y = sY[c1 * 4 + q1];
    b2v.x = sY[c2 * 4 + q0]; b2v.y = sY[c2 * 4 + q1];
    b3.x  = sY[c3 * 4 + q0]; b3.y  = sY[c3 * 4 + q1];
    v8f zero = {};
    v8f d0 = wmma_k4(a, b0,  zero);
    v8f d1 = wmma_k4(a, b1v, zero);
    v8f d2 = wmma_k4(a, b2v, zero);
    v8f d3 = wmma_k4(a, b3,  zero);
    const float y0 = sYsq[c0], y1 = sYsq[c1], y2 = sYsq[c2], y3 = sYsq[c3];
    const float t0 = sCv[c0],  t1 = sCv[c1],  t2 = sCv[c2],  t3 = sCv[c3];
#pragma unroll
    for (int i = 0; i < 8; ++i) {
      float a0 = alpha2 * fmaxf(xsqv[i] + y0 - 2.0f * d0[i], 0.0f) + t0;
      float a1 = alpha2 * fmaxf(xsqv[i] + y1 - 2.0f * d1[i], 0.0f) + t1;
      float a2 = alpha2 * fmaxf(xsqv[i] + y2 - 2.0f * d2[i], 0.0f) + t2;
      float a3 = alpha2 * fmaxf(xsqv[i] + y3 - 2.0f * d3[i], 0.0f) + t3;
      float tm = fmaxf(fmaxf(a0, a1), fmaxf(a2, a3));
      float ts = exp2f(a0 - tm) + exp2f(a1 - tm) +
                 exp2f(a2 - tm) + exp2f(a3 - tm);
      lse2_accum(mm[i], ss[i], tm, ts);
    }
  }
#pragma unroll
  for (int i = 0; i < 8; ++i) {                // reduce the 16 cols/lane-group
    for (int off = 1; off < 16; off <<= 1) {
      float m2 = __shfl_xor(mm[i], off, 16);
      float s2 = __shfl_xor(ss[i], off, 16);
      lse2_merge(mm[i], ss[i], m2, s2);
    }
  }
  if (li == 0) {
#pragma unroll
    for (int i = 0; i < 8; ++i)
      out[rowbase + i + (hi ? 8 : 0)] = rscaleLn2 * (mm[i] + log2f(ss[i]));
  }
}

// --------------------------------------------- generic logsumexp col sweep
// out[m] = rscaleLn2 * lse2_n( alpha2*dist(X[n],Y[m]) + beta2*rowv[n] + gamma2 )
__global__ __launch_bounds__(256) void lse_col_pass(
    const float* __restrict__ Xpts, const float* __restrict__ Xsq,
    int xPtsStride, int xSqStride,
    const float* __restrict__ Ypts, const float* __restrict__ Ysq,
    const float* __restrict__ rowv, float* __restrict__ outCol,
    float alpha2, float beta2, float gamma2, float rscaleLn2) {
  __shared__ __align__(16) float sX[NSUB * 4];
  __shared__ float sXsq[NSUB];
  __shared__ float sRv[NSUB];    // beta2*rowv + gamma2 (pre-folded)
  const int b = blockIdx.y;
  const float* xp = Xpts + (long)b * xPtsStride;
  const float* xs = Xsq  + (long)b * xSqStride;
  const float* yp = Ypts + (long)b * (NSUB * 4);
  const float* ys = Ysq  + (long)b * NSUB;
  const float* rv = rowv + (long)b * NSUB;
  float* out = outCol + (long)b * NSUB;
  const int tid = threadIdx.x;
  for (int i = tid; i < NSUB; i += 256) {
    ((float4*)sX)[i] = ((const float4*)xp)[i];
    sXsq[i] = xs[i];
    sRv[i]  = beta2 * rv[i] + gamma2;
  }
  __syncthreads();
  const int wave = tid >> 5, lane = tid & 31, li = lane & 15;
  const bool hi = lane >= 16;
  const int q0 = hi ? 2 : 0, q1 = hi ? 3 : 1;
  const int colbase = (blockIdx.x * 8 + wave) * 16;
  const int col = colbase + li;

  v2f bb;                                      // fixed B: 16 cols of Y
  bb.x = yp[col * 4 + q0];
  bb.y = yp[col * 4 + q1];
  const float ysqc = ys[col];

  float m = NEG_INF, s = 0.0f;
  for (int t = 0; t < NSUB / 16; ++t) {
    v2f a;
    a.x = sX[(t * 16 + li) * 4 + q0];
    a.y = sX[(t * 16 + li) * 4 + q1];
    v8f zero = {};
    v8f d = wmma_k4(a, bb, zero);
    const int rb = t * 16 + (hi ? 8 : 0);
    float args[8];
#pragma unroll
    for (int i = 0; i < 8; ++i) {
      float dist = fmaxf(sXsq[rb + i] + ysqc - 2.0f * d[i], 0.0f);
      args[i] = alpha2 * dist + sRv[rb + i];
    }
    // 10 exps per 8 elements: tile max + batched rescale
    float tm = fmaxf(fmaxf(fmaxf(args[0], args[1]), fmaxf(args[2], args[3])),
                     fmaxf(fmaxf(args[4], args[5]), fmaxf(args[6], args[7])));
    float ts = exp2f(args[0] - tm) + exp2f(args[1] - tm) +
               exp2f(args[2] - tm) + exp2f(args[3] - tm) +
               exp2f(args[4] - tm) + exp2f(args[5] - tm) +
               exp2f(args[6] - tm) + exp2f(args[7] - tm);
    lse2_accum(m, s, tm, ts);
  }
  {                                            // halves hold the same column
    float m2 = __shfl_xor(m, 16, 32);
    float s2 = __shfl_xor(s, 16, 32);
    lse2_merge(m, s, m2, s2);
  }
  if (!hi) out[col] = rscaleLn2 * (m + log2f(s));
}

// ---------------------------------------------- target_perm = P @ target_sub
__global__ __launch_bounds__(256) void perm_kernel(
    const float* __restrict__ tpts, const float* __restrict__ tsq,
    const float* __restrict__ ypts, const float* __restrict__ ysq,
    const float* __restrict__ lu, const float* __restrict__ lv,
    float* __restrict__ permp) {
  __shared__ __align__(16) float sY[NSUB * 4];
  __shared__ float sYsq[NSUB];
  __shared__ float sLv[NSUB];
  const int b = blockIdx.y, tid = threadIdx.x;
  const float* yp = ypts + (long)b * NSUB * 4;
  const float* ys = ysq + (long)b * NSUB;
  const float* lvb = lv + (long)b * NSUB;
  for (int i = tid; i < NSUB; i += 256) {
    ((float4*)sY)[i] = ((const float4*)yp)[i];
    sYsq[i] = ys[i];
    sLv[i] = LOG2E * lvb[i];                   // base-2 pre-scale
  }
  __syncthreads();
  const int wave = tid >> 5, lane = tid & 31;
  const int row = blockIdx.x * 8 + wave;
  const float a2 = -LOG2E / S_REG;
  const float tx = tpts[row * 4 + 0], ty = tpts[row * 4 + 1], tz = tpts[row * 4 + 2];
  const float tq = tsq[row];
  const float lur2 = LOG2E * lu[(long)b * NSUB + row];
  float ax = 0.0f, ay = 0.0f, az = 0.0f;
  for (int mcol = lane; mcol < NSUB; mcol += 32) {
    float yx = sY[mcol * 4 + 0], yy = sY[mcol * 4 + 1], yz = sY[mcol * 4 + 2];
    float dist = fmaxf(tq + sYsq[mcol] - 2.0f * (tx * yx + ty * yy + tz * yz), 0.0f);
    float p = exp2f(a2 * dist + lur2 + sLv[mcol]);
    ax += p * yx; ay += p * yy; az += p * yz;
  }
  for (int off = 1; off < 32; off <<= 1) {
    ax += __shfl_xor(ax, off, 32);
    ay += __shfl_xor(ay, off, 32);
    az += __shfl_xor(az, off, 32);
  }
  if (lane == 0) {
    long o = ((long)b * NSUB + row) * 4;
    permp[o + 0] = ax; permp[o + 1] = ay; permp[o + 2] = az; permp[o + 3] = 0.0f;
  }
}

// -------------------------------- z = max_n relu(tsub @ W_enc + b_enc), WMMA
__global__ __launch_bounds__(256) void feat_z_kernel(
    const float* __restrict__ tsubp, const float* __restrict__ Wenc,
    const float* __restrict__ benc, float* __restrict__ zbuf) {
  const int b = blockIdx.y, tid = threadIdx.x;
  const int wave = tid >> 5, lane = tid & 31, li = lane & 15;
  const bool hi = lane >= 16;
  const int rowbase = blockIdx.x * 16;
  const float* tp = tsubp + (long)b * NSUB * 4;
  v2f a;
  a.x = tp[(rowbase + li) * 4 + (hi ? 2 : 0)];
  a.y = tp[(rowbase + li) * 4 + (hi ? 3 : 1)];
  const int k0 = hi ? 2 : 0, k1 = hi ? 3 : 1;
  for (int j = wave * 2; j < wave * 2 + 2; ++j) {
    int col = j * 16 + li;
    v2f bb;
    bb.x = (k0 < 3) ? Wenc[k0 * DZ + col] : 0.0f;
    bb.y = (k1 < 3) ? Wenc[k1 * DZ + col] : 0.0f;
    v8f zero = {};
    v8f d = wmma_k4(a, bb, zero);
    float bias = benc[col];
    float mx = 0.0f;                            // relu floor
#pragma unroll
    for (int i = 0; i < 8; ++i) mx = fmaxf(mx, d[i] + bias);
    float o = __shfl_xor(mx, 16, 32);
    mx = fmaxf(mx, o);
    if (!hi)  // nonneg floats: IEEE order == uint order
      atomicMax((unsigned int*)&zbuf[(long)b * DZ + col], __float_as_uint(mx));
  }
}

// --------------- v_pred MLP (WMMA), loss_velocity partials, x1 = tmpl+v_pred
__global__ __launch_bounds__(256) void mlp_kernel(
    const float* __restrict__ tpts, const float* __restrict__ permp,
    const float* __restrict__ tvals, const float* __restrict__ zbuf,
    const float* __restrict__ W1, const float* __restrict__ b1,
    const float* __restrict__ W2, const float* __restrict__ b2,
    float* __restrict__ x1p, float* __restrict__ x1sq,
    float* __restrict__ accs) {
  __shared__ float hin[16 * DIN_PAD];
  __shared__ float h2[16 * DH];
  const int b = blockIdx.y, tid = threadIdx.x;
  const int rowbase = blockIdx.x * 16;
  const float tc = fminf(fmaxf(tvals[b], 1e-5f), 1.0f - 1e-5f);
  const float* pm = permp + (long)b * NSUB * 4;
  for (int idx = tid; idx < 16 * DIN_PAD; idx += 256) {
    int r = idx / DIN_PAD, c = idx % DIN_PAD;
    int row = rowbase + r;
    float v;
    if (c < 3)        v = (1.0f - tc) * tpts[row * 4 + c] + tc * pm[row * 4 + c];
    else if (c == 3)  v = tc;
    else if (c < DIN) v = zbuf[(long)b * DZ + (c - 4)];
    else              v = 0.0f;
    hin[idx] = v;
  }
  __syncthreads();
  const int wave = tid >> 5, lane = tid & 31, li = lane & 15;
  const bool hi = lane >= 16;
  const int q0 = hi ? 2 : 0, q1 = hi ? 3 : 1;
  for (int j = wave * 2; j < wave * 2 + 2; ++j) {      // layer 1: K=260
    int col = j * 16 + li;
    v8f acc = {};
    for (int k = 0; k < DIN_PAD / 4; ++k) {
      int kk = k * 4;
      v2f a; a.x = hin[li * DIN_PAD + kk + q0]; a.y = hin[li * DIN_PAD + kk + q1];
      v2f bb;
      bb.x = (kk + q0 < DIN) ? W1[(kk + q0) * DH + col] : 0.0f;
      bb.y = (kk + q1 < DIN) ? W1[(kk + q1) * DH + col] : 0.0f;
      acc = wmma_k4(a, bb, acc);
    }
    float bias = b1[col];
#pragma unroll
    for (int i = 0; i < 8; ++i)
      h2[(i + (hi ? 8 : 0)) * DH + col] = fmaxf(acc[i] + bias, 0.0f);
  }
  __syncthreads();
  if (wave == 0) {                                     // layer 2: K=256, N=3
    v8f acc = {};
    for (int k = 0; k < DH / 4; ++k) {
      int kk = k * 4;
      v2f a; a.x = h2[li * DH + kk + q0]; a.y = h2[li * DH + kk + q1];
      v2f bb;
      bb.x = (li < 3) ? W2[(kk + q0) * 3 + li] : 0.0f;
      bb.y = (li < 3) ? W2[(kk + q1) * 3 + li] : 0.0f;
      acc = wmma_k4(a, bb, acc);
    }
    float velsum = 0.0f;
    float* xb  = x1p  + (long)b * NSUB * 4;
    float* xsq = x1sq + (long)b * NSUB;
#pragma unroll
    for (int i = 0; i < 8; ++i) {
      int row = rowbase + i + (hi ? 8 : 0);
      float x1v = 0.0f;
      if (li < 3) {
        float vp = acc[i] + b2[li];
        float tv = tpts[row * 4 + li];
        float vgt = pm[row * 4 + li] - tv;
        float df = vp - vgt;
        velsum += df * df;
        x1v = tv + vp;
        xb[row * 4 + li] = x1v;
      } else if (li == 3) {
        xb[row * 4 + 3] = 0.0f;                        // K-pad must be zero
      }
      float sq = x1v * x1v;                            // zero for li>=3
      for (int off = 1; off < 16; off <<= 1) sq += __shfl_xor(sq, off, 16);
      if (li == 0) xsq[row] = sq;
    }
    for (int off = 1; off < 32; off <<= 1) velsum += __shfl_xor(velsum, off, 32);
    if (lane == 0) atomicAdd(&accs[0], velsum);
  }
}

// ----------------------------------------------------------- OT reductions
__global__ __launch_bounds__(256) void ot_reduce_kernel(
    const float* __restrict__ f, const float* __restrict__ g,
    float* __restrict__ otpb) {
  __shared__ float sr[256];
  const int p = blockIdx.x, b = blockIdx.y, tid = threadIdx.x;
  const float* fp = f + ((long)p * B_SIZE + b) * NSUB;
  const float* gp = g + ((long)p * B_SIZE + b) * NSUB;
  float s = 0.0f;
  for (int i = tid; i < NSUB; i += 256) s += fp[i] + gp[i];
  sr[tid] = s;
  __syncthreads();
  for (int off = 128; off > 0; off >>= 1) {
    if (tid < off) sr[tid] += sr[tid + off];
    __syncthreads();
  }
  if (tid == 0) otpb[p * B_SIZE + b] = sr[0] * (1.0f / NSUB);
}

__global__ void final_kernel(const float* __restrict__ otpb,
                             const float* __restrict__ accs,
                             float* __restrict__ out) {
  if (threadIdx.x == 0 && blockIdx.x == 0) {
    float lot = 0.0f;
    for (int b = 0; b < B_SIZE; ++b)
      lot += otpb[0 * B_SIZE + b] - 0.5f * otpb[1 * B_SIZE + b]
                                  - 0.5f * otpb[2 * B_SIZE + b];
    lot *= (1.0f / B_SIZE);
    float lvel = accs[0] / (float)(B_SIZE * NSUB * 3);
    float lreg = accs[1] / (float)NSUB;
    out[0] = lvel + 0.1f * lot + 0.01f * lreg;
    out[1] = lvel;
    out[2] = lot;
    out[3] = lreg;
  }
}

// ------------------------------------------------------------------- driver
extern "C" void kernel_launch(void* const* d_in, const int* in_sizes, int n_in,
                              void* d_out, int out_size, void* d_ws, size_t ws_size,
                              hipStream_t stream) {
  (void)in_sizes; (void)n_in; (void)out_size; (void)ws_size;
  const float* target = (const float*)d_in[0];
  const float* tvals  = (const float*)d_in[1];
  const float* tmpl   = (const float*)d_in[2];
  const float* Wenc   = (const float*)d_in[3];
  const float* benc   = (const float*)d_in[4];
  const float* W1     = (const float*)d_in[5];
  const float* b1     = (const float*)d_in[6];
  const float* W2     = (const float*)d_in[7];
  const float* b2     = (const float*)d_in[8];
  float* out = (float*)d_out;
  float* w = (float*)d_ws;

  // ---- workspace layout (floats); zero-region first for one memset
  float* lu     = w;                  // 8*2048
  float* lv     = lu + 16384;         // 8*2048
  float* fbuf   = lv + 16384;         // 3*8*2048
  float* gbuf   = fbuf + 49152;       // 3*8*2048
  float* zbuf   = gbuf + 49152;       // 8*256
  float* accs   = zbuf + 2048;        // [0]=vel_sum, [1]=reg_sum
  const size_t zeroFloats = 16384 + 16384 + 49152 + 49152 + 2048 + 16;
  float* tpts   = accs + 16;          // 2048*4  (xyz,0)
  float* tsq    = tpts + 8192;        // 2048
  float* tsubp  = tsq + 2048;         // 8*2048*4
  float* tsubsq = tsubp + 65536;      // 8*2048
  float* permp  = tsubsq + 16384;     // 8*2048*4
  float* x1p    = permp + 65536;      // 8*2048*4
  float* x1sq   = x1p + 65536;        // 8*2048
  float* otpb   = x1sq + 16384;       // 3*8

  hipMemsetAsync(d_ws, 0, zeroFloats * sizeof(float), stream);

  prep_template_kernel<<<8, 256, 0, stream>>>(tmpl, tpts, tsq, accs);
  fps_kernel<<<B_SIZE, 1024, 0, stream>>>(target, tsubp, tsubsq);

  // ---- Sinkhorn assignment (base-2): arg = -dist*log2e/reg + log2e*lv[m]
  const float aS2 = -LOG2E / S_REG, bS2 = LOG2E, gS2 = 0.0f, rS2 = -LN2;
  dim3 pg(16, B_SIZE);
  for (int it = 0; it < S_ITERS; ++it) {
    lse_row_pass<<<pg, 256, 0, stream>>>(tpts, tsq, 0, 0, tsubp, tsubsq,
                                         lv, lu, aS2, bS2, gS2, rS2);
    lse_col_pass<<<pg, 256, 0, stream>>>(tpts, tsq, 0, 0, tsubp, tsubsq,
                                         lu, lv, aS2, bS2, gS2, rS2);
  }
  perm_kernel<<<dim3(256, B_SIZE), 256, 0, stream>>>(tpts, tsq, tsubp, tsubsq,
                                                     lu, lv, permp);
  feat_z_kernel<<<dim3(128, B_SIZE), 256, 0, stream>>>(tsubp, Wenc, benc, zbuf);
  mlp_kernel<<<dim3(128, B_SIZE), 256, 0, stream>>>(tpts, permp, tvals, zbuf,
                                                    W1, b1, W2, b2,
                                                    x1p, x1sq, accs);

  // ---- Sinkhorn divergence: 3 OT problems, f/g updates (base-2)
  // arg = (g[m] - 0.5*dist)/eps + log_nu, all scaled by log2e
  const float aO2 = -0.5f * LOG2E / OT_EPS;
  const float bO2 = LOG2E / OT_EPS;
  const float gO2 = LOG_NU * LOG2E;
  const float rO2 = -OT_EPS * LN2;
  const float* XP[3] = { x1p,    x1p,  tsubp };
  const float* XS[3] = { x1sq,   x1sq, tsubsq };
  const float* YP[3] = { tsubp,  x1p,  tsubp };
  const float* YS[3] = { tsubsq, x1sq, tsubsq };
  for (int it = 0; it < OT_ITERS; ++it) {
    for (int p = 0; p < 3; ++p)
      lse_row_pass<<<pg, 256, 0, stream>>>(XP[p], XS[p], NSUB * 4, NSUB,
                                           YP[p], YS[p],
                                           gbuf + p * 16384, fbuf + p * 16384,
                                           aO2, bO2, gO2, rO2);
    for (int p = 0; p < 3; ++p)
      lse_col_pass<<<pg, 256, 0, stream>>>(XP[p], XS[p], NSUB * 4, NSUB,
                                           YP[p], YS[p],
                                           fbuf + p * 16384, gbuf + p * 16384,
                                           aO2, bO2, gO2, rO2);
  }
  ot_reduce_kernel<<<dim3(3, B_SIZE), 256, 0, stream>>>(fbuf, gbuf, otpb);
  final_kernel<<<1, 32, 0, stream>>>(otpb, accs, out);
}